// CurvatureEstimator_60876866454092
// MI455X (gfx1250) — compile-verified
//
#include <hip/hip_runtime.h>

// CurvatureEstimator fused kernel for gfx1250 (MI455X).
// B=16, H=512, W=512, C=8, fp32. Separable 11x11 stencil:
//   pass1 (W): S0 = sum e, S1 = sum dj*e, S2 = sum dj^2*e   (float4 VALU out of LDS)
//   pass2 (H): banded 16x28 matmul via V_WMMA_F32_16X16X4_F32:
//       ox = lin*S0, sx = sqr*S0, oy = box*S1, sy = box*S2
//   out = 2*sqrt(ox^2+oy^2)/(sx+sy+1e-5)
// Step-1 tile fill uses GLOBAL_LOAD_ASYNC_TO_LDS_B128 (ASYNCcnt) — no VGPR staging.
// WMMA A-operand coefficients come from a compile-time __constant__ table.

typedef __attribute__((ext_vector_type(2))) float v2f;
typedef __attribute__((ext_vector_type(8))) float v8f;

#define Bn 16
#define Hn 512
#define Wn 512
#define Cn 8
#define TH 16          // output rows per block
#define TW 8           // output pixels per block
#define TWC (TW * Cn)  // 64 output columns (w,c pairs)
#define IN_ROWS (TH + 10)      // 26
#define IN_PIX  (TW + 10)      // 18
#define S_ROWS  (IN_ROWS + 2)  // 28: pad K dim to multiple of 4
#define S_PITCH 72             // (2*72)%64 = 16 -> lane halves hit disjoint banks

// ---- Compile-time banded-coefficient table, laid out per WMMA lane striping ----
// A 16x4 tile: lane holds M = lane&15; vgpr v holds K = 2*(lane>>4)+v.
// Chunk q covers K-matrix columns r = 4q..4q+3; K[o][r] = w[r-o] for 0<=r-o<=10.
struct ATable { float v[32][3][16]; };   // [lane][lin|sqr|box][2*q+vv], 6 KB
static constexpr ATable makeATable() {
    ATable t = {};
    for (int lane = 0; lane < 32; ++lane) {
        int m = lane & 15, half = lane >> 4;
        for (int q = 0; q < 7; ++q)
            for (int vv = 0; vv < 2; ++vv) {
                int j = 4 * q + 2 * half + vv - m;   // tap index r - o
                bool ok = (j >= 0) && (j <= 10);
                float fl = ok ? (float)(j - 5) : 0.0f;
                t.v[lane][0][2 * q + vv] = fl;            // linear tap (di)
                t.v[lane][1][2 * q + vv] = fl * fl;       // di^2
                t.v[lane][2][2 * q + vv] = ok ? 1.f : 0.f; // box tap
            }
    }
    return t;
}
__constant__ ATable gATable = makeATable();

__device__ __forceinline__ void fma4(float a, const float4& x, float4& acc) {
    acc.x = fmaf(a, x.x, acc.x);
    acc.y = fmaf(a, x.y, acc.y);
    acc.z = fmaf(a, x.z, acc.z);
    acc.w = fmaf(a, x.w, acc.w);
}

__launch_bounds__(128)
__global__ void curvature_wmma_kernel(const float* __restrict__ in,
                                      float* __restrict__ out)
{
    const int b   = blockIdx.z;
    const int h0  = blockIdx.y * TH;
    const int w0  = blockIdx.x * TW;
    const int tid = threadIdx.x;   // 0..127, 4 waves of 32

    __shared__ __align__(16) float lds_in[IN_ROWS][IN_PIX * Cn];  // 26 x 144 = 14976 B
    __shared__ __align__(16) float sS[3][S_ROWS][S_PITCH];        // 3 x 28 x 72 = 24192 B

    // ---- Step 1: async global -> LDS (b128 per lane), zero-fill OOB halo ----
    for (int i = tid; i < IN_ROWS * (IN_PIX * 2); i += 128) {
        int r  = i / (IN_PIX * 2);
        int q4 = i - r * (IN_PIX * 2);   // float4 slot: pixel p = q4>>1, channel half = q4&1
        int p  = q4 >> 1;
        int h  = h0 + r - 5;
        int w  = w0 + p - 5;
        float* lp = &lds_in[r][q4 * 4];
        if ((unsigned)h < (unsigned)Hn && (unsigned)w < (unsigned)Wn) {
            const float* gp = in + ((((size_t)b * Hn + h) * Wn + w) * Cn) + (q4 & 1) * 4;
            unsigned ldsoff = (unsigned)(unsigned long long)lp;  // wave-relative LDS address
            asm volatile("global_load_async_to_lds_b128 %0, %1, off"
                         :: "v"(ldsoff), "v"(gp) : "memory");
        } else {
            *reinterpret_cast<float4*>(lp) = make_float4(0.f, 0.f, 0.f, 0.f);
        }
    }

    // Fetch this lane's WMMA coefficients (L2-resident constant table) while
    // the async engine fills LDS.
    const int lane = tid & 31;
    const v2f* Alin = reinterpret_cast<const v2f*>(gATable.v[lane][0]);
    const v2f* Asqr = reinterpret_cast<const v2f*>(gATable.v[lane][1]);
    const v2f* Abox = reinterpret_cast<const v2f*>(gATable.v[lane][2]);

#if __has_builtin(__builtin_amdgcn_s_wait_asynccnt)
    __builtin_amdgcn_s_wait_asynccnt(0);
#else
    asm volatile("s_wait_asynccnt 0" ::: "memory");
#endif
    __syncthreads();

    // ---- Step 2: horizontal 11-tap pass, float4 over channels -> S0,S1,S2 ----
    for (int i = tid; i < IN_ROWS * (TW * 2); i += 128) {  // 416 float4 columns
        int r    = i >> 4;
        int c4   = i & 15;
        int base = c4 * 4;              // float offset in row (= w*8 + chalf*4)
        float4 s0 = make_float4(0.f, 0.f, 0.f, 0.f);
        float4 s1 = s0, s2 = s0;
#pragma unroll
        for (int t = 0; t < 11; ++t) {
            float4 x = *reinterpret_cast<const float4*>(&lds_in[r][base + t * Cn]);
            float dj = (float)(t - 5);
            s0.x += x.x; s0.y += x.y; s0.z += x.z; s0.w += x.w;
            fma4(dj, x, s1);
            fma4(dj * dj, x, s2);
        }
        *reinterpret_cast<float4*>(&sS[0][r][base]) = s0;
        *reinterpret_cast<float4*>(&sS[1][r][base]) = s1;
        *reinterpret_cast<float4*>(&sS[2][r][base]) = s2;
    }
    // zero the two K-padding rows (26, 27) of each S array (96 float4s)
    for (int i = tid; i < 3 * 2 * (TWC / 4); i += 128) {
        int s   = i / 32;
        int rem = i - s * 32;
        int r   = IN_ROWS + (rem >> 4);
        int c4  = (rem & 15) * 4;
        *reinterpret_cast<float4*>(&sS[s][r][c4]) = make_float4(0.f, 0.f, 0.f, 0.f);
    }
    __syncthreads();

    // ---- Step 3: vertical banded matmul via V_WMMA_F32_16X16X4_F32 ----
    // out16x16 = K(16x28) * S(28x16).
    //   B 4x16 : lane holds N = lane&15; vgpr v holds K = 2*(lane>>4)+v
    //   C/D    : lane holds N = lane&15; vgpr r holds M = r + 8*(lane>>4)
    const int grp  = tid >> 5;       // one 16-column group per wave
    const int n    = lane & 15;
    const int half = lane >> 4;
    const int colg = grp * 16 + n;

    v8f ox = {}, oy = {}, sx = {}, sy = {};
#pragma unroll
    for (int q = 0; q < 7; ++q) {
        const int r0 = 4 * q + 2 * half;   // S row for B vgpr0; vgpr1 -> r0+1
        v2f b0, b1, b2;
        b0.x = sS[0][r0][colg]; b0.y = sS[0][r0 + 1][colg];
        b1.x = sS[1][r0][colg]; b1.y = sS[1][r0 + 1][colg];
        b2.x = sS[2][r0][colg]; b2.y = sS[2][r0 + 1][colg];

        ox = __builtin_amdgcn_wmma_f32_16x16x4_f32(false, Alin[q], false, b0, (short)0, ox, false, false);
        sx = __builtin_amdgcn_wmma_f32_16x16x4_f32(false, Asqr[q], false, b0, (short)0, sx, false, false);
        oy = __builtin_amdgcn_wmma_f32_16x16x4_f32(false, Abox[q], false, b1, (short)0, oy, false, false);
        sy = __builtin_amdgcn_wmma_f32_16x16x4_f32(false, Abox[q], false, b2, (short)0, sy, false, false);
    }

    // ---- Step 4: combine + store (coalesced 16-lane row segments) ----
    float* obase = out + (((size_t)b * Hn + h0) * Wn + w0) * Cn + (size_t)colg;
#pragma unroll
    for (int r = 0; r < 8; ++r) {
        int mm = r + 8 * half;
        float vox = ox[r], voy = oy[r], vsx = sx[r], vsy = sy[r];
        float res = 2.0f * sqrtf(vox * vox + voy * voy) / (vsx + vsy + 1e-5f);
        obase[(size_t)mm * (Wn * Cn)] = res;
    }
}

extern "C" void kernel_launch(void* const* d_in, const int* in_sizes, int n_in,
                              void* d_out, int out_size, void* d_ws, size_t ws_size,
                              hipStream_t stream) {
    const float* edges = (const float*)d_in[0];
    float* out = (float*)d_out;
    dim3 grid(Wn / TW, Hn / TH, Bn);   // 64 x 32 x 16 = 32768 blocks
    dim3 block(128);
    curvature_wmma_kernel<<<grid, block, 0, stream>>>(edges, out);
}